// DHGCN_DGCNN_53893249630260
// MI455X (gfx1250) — compile-verified
//
#include <hip/hip_runtime.h>
#include <math.h>

typedef float v2f __attribute__((ext_vector_type(2)));
typedef float v8f __attribute__((ext_vector_type(8)));

#define NPTS 1024
#define BATCH 16
#define KNN 20
#define VP 27
#define ESTR 736      // padded 27*27=729 row stride for WMMA-friendly tiles
#define FEATC 512

__device__ __forceinline__ float lrelu(float z) { return z > 0.f ? z : 0.2f * z; }

// CDNA5 fp32 WMMA: D = A(16x4) * B(4x16) + C(16x16), wave32, one tile per wave
__device__ __forceinline__ v8f wmma4(v2f a, v2f b, v8f c) {
  return __builtin_amdgcn_wmma_f32_16x16x4_f32(false, a, false, b, (short)0, c,
                                               false, false);
}

// ---------------------------------------------------------------------------
// Tensor Data Mover: stage a 16-row x C-channel strip (stride NPTS) into LDS.
#if defined(__AMDGCN__) && __has_builtin(__builtin_amdgcn_tensor_load_to_lds)
#define USE_TDM 1
#else
#define USE_TDM 0
#endif

#if USE_TDM
typedef unsigned tdm_v4u __attribute__((ext_vector_type(4)));
typedef int tdm_v4i __attribute__((ext_vector_type(4)));
typedef int tdm_v8i __attribute__((ext_vector_type(8)));

__device__ __forceinline__ void tdm_load_strip(const float* gaddr,
                                               unsigned lds_off, int C) {
  unsigned long long ga = (unsigned long long)gaddr;
  tdm_v4u g0;
  g0[0] = 1u;                                   // count=1, user descriptor
  g0[1] = lds_off;                              // lds_addr (bytes)
  g0[2] = (unsigned)(ga & 0xFFFFFFFFu);         // global_addr[31:0]
  g0[3] = (unsigned)((ga >> 32) & 0x1FFFFFFu)   // global_addr[56:32]
          | (2u << 30);                         // type=2 ("image")
  tdm_v8i g1;
  g1[0] = (int)(2u << 16);                      // data_size=4B, no multicast
  g1[1] = (int)((unsigned)(NPTS & 0xFFFF) << 16);       // tensor_dim0 lo16
  g1[2] = (int)((unsigned)(C & 0xFFFF) << 16);          // dim0 hi=0 | dim1 lo=C
  g1[3] = (int)(16u << 16);                     // dim1 hi=0 | tile_dim0=16
  g1[4] = C;                                    // tile_dim1=C, tile_dim2=0
  g1[5] = NPTS;                                 // tensor_dim0_stride lo32
  g1[6] = 0;                                    // stride0 hi | stride1 lo
  g1[7] = 0;
  tdm_v4i z4 = {};
#if __clang_major__ >= 23
  tdm_v8i z8 = {};
  __builtin_amdgcn_tensor_load_to_lds(g0, g1, z4, z4, z8, 0);
#else
  __builtin_amdgcn_tensor_load_to_lds(g0, g1, z4, z4, 0);
#endif
}
#endif

// ---------------------------------------------------------------------------
// Part bookkeeping: invert part_rand permutation -> per-point part column.
__global__ void init_parts_k(const int* p2v, const int* prand, int* pcol,
                             float* cnt) {
  int b = blockIdx.x, lane = threadIdx.x;
  __shared__ int inv[VP];
  __shared__ int cbin[VP];
  if (lane < VP) { inv[prand[b * VP + lane]] = lane; cbin[lane] = 0; }
  __syncthreads();
  for (int n = lane; n < NPTS; n += 32) {
    int p = inv[p2v[b * NPTS + n]];
    pcol[b * NPTS + n] = p;
    atomicAdd(&cbin[p], 1);
  }
  __syncthreads();
  if (lane < VP) cnt[b * 32 + lane] = fmaxf((float)cbin[lane], 1.f);
}

// pce = lrelu(s * (Wpc . part_center) + b), 64 x 27 per batch (tiny)
__global__ void pce_k(const float* x, const int* pcol, const float* cnt,
                      const float* w, const float* s, const float* bb,
                      float* pce) {
  int b = blockIdx.x, lane = threadIdx.x;
  if (lane >= VP) return;
  float a0 = 0.f, a1 = 0.f, a2 = 0.f;
  for (int n = 0; n < NPTS; n++) {
    if (pcol[b * NPTS + n] == lane) {
      a0 += x[(b * 3 + 0) * NPTS + n];
      a1 += x[(b * 3 + 1) * NPTS + n];
      a2 += x[(b * 3 + 2) * NPTS + n];
    }
  }
  float c = cnt[b * 32 + lane];
  a0 /= c; a1 /= c; a2 /= c;
  for (int o = 0; o < 64; o++) {
    float z = w[o * 3] * a0 + w[o * 3 + 1] * a1 + w[o * 3 + 2] * a2;
    pce[(b * 64 + o) * 32 + lane] = lrelu(s[o] * z + bb[o]);
  }
}

// per-column squared norms (the -xx[m] term; -xx[n] is rank-invariant per row)
__global__ void colxx_k(const float* h, long hbs, int C, float* colxx) {
  int b = blockIdx.y;
  int n = blockIdx.x * blockDim.x + threadIdx.x;
  const float* hb = h + (long)b * hbs;
  float s = 0.f;
  for (int c = 0; c < C; c++) { float v = hb[c * NPTS + n]; s += v * v; }
  colxx[b * NPTS + n] = s;
}

// ---------------------------------------------------------------------------
// Fused kNN: TDM stages the 16-row A-strip into LDS; one wave computes a
// 16x1024 Gram strip via WMMA f32 16x16x4 entirely in LDS, then 20-pass
// max-selection per row. The NxN distance matrix never touches HBM.
__global__ void knn_k(const float* h, long hbs, int C, const float* colxx,
                      int* idx) {
  __shared__ float sd[16 * NPTS];  // 64 KB score strip
  __shared__ float sA[16 * 128];   // A strip: [c][row], row fastest (16 floats per c)
  int b = blockIdx.y;
  int rowbase = blockIdx.x * 16;
  int lane = threadIdx.x;
  int m16 = lane & 15, hi = lane >> 4;
  const float* hb = h + (long)b * hbs;

#if USE_TDM
  tdm_load_strip(hb + rowbase, (unsigned)(unsigned long long)&sA[0], C);
  __builtin_amdgcn_s_wait_tensorcnt(0);
#else
  for (int t = lane; t < C * 16; t += 32)
    sA[t] = hb[(t >> 4) * NPTS + rowbase + (t & 15)];
  __syncthreads();
#endif

  int C4 = C & ~3;
  for (int t = 0; t < NPTS / 16; t++) {
    int colbase = t * 16;
    int col = colbase + m16;
    v8f acc = {};
#pragma unroll 2
    for (int c0 = 0; c0 < C4; c0 += 4) {
      int ca = c0 + 2 * hi;
      v2f a, bf;
      a.x = sA[ca * 16 + m16];
      a.y = sA[(ca + 1) * 16 + m16];
      bf.x = hb[ca * NPTS + col];
      bf.y = hb[(ca + 1) * NPTS + col];
      acc = wmma4(a, bf, acc);
    }
    if (C & 3) {  // C==3 tail (layer 0 only)
      int ca = C4 + 2 * hi;
      v2f a, bf;
      a.x  = (ca     < C) ? sA[ca * 16 + m16]       : 0.f;
      a.y  = (ca + 1 < C) ? sA[(ca + 1) * 16 + m16] : 0.f;
      bf.x = (ca     < C) ? hb[ca * NPTS + col]       : 0.f;
      bf.y = (ca + 1 < C) ? hb[(ca + 1) * NPTS + col] : 0.f;
      acc = wmma4(a, bf, acc);
    }
    float cxx = colxx[b * NPTS + col];
#pragma unroll
    for (int r = 0; r < 8; r++) {
      int mm = r + 8 * hi;  // C/D layout: lanes 0-15 -> M=r, lanes 16-31 -> M=r+8
      sd[mm * NPTS + col] = 2.f * acc[r] - cxx;
    }
  }
  __syncthreads();
  if (lane < 16) {
    for (int kk = 0; kk < KNN; kk++) {
      float best = -3.0e38f; int bi = 0;
      for (int j = 0; j < NPTS; j++) {
        float v = sd[lane * NPTS + j];
        if (v > best) { best = v; bi = j; }
      }
      idx[(b * NPTS + rowbase + lane) * KNN + kk] = bi;
      sd[lane * NPTS + bi] = -3.4e38f;
    }
  }
}

// ---------------------------------------------------------------------------
// Edge-conv decomposition GEMMs: A = W1.h, D = (W2-W1).h, both via WMMA.
__global__ void gemmAD_k(const float* h, long hbs, int C, int twoC,
                         const float* w, float* A, float* D) {
  int b = blockIdx.z;
  int obase = blockIdx.y * 16;
  int nbase = blockIdx.x * 16;
  int lane = threadIdx.x, m16 = lane & 15, hi = lane >> 4;
  const float* hb = h + (long)b * hbs;
  int C4 = C & ~3;
  v8f accA = {}, accD = {};
  int orow = obase + m16;
  int ncol = nbase + m16;
#pragma unroll 2
  for (int c0 = 0; c0 < C4; c0 += 4) {
    int ca = c0 + 2 * hi;
    float w1a = w[orow * twoC + ca];
    float w1b = w[orow * twoC + ca + 1];
    float w2a = w[orow * twoC + C + ca];
    float w2b = w[orow * twoC + C + ca + 1];
    v2f a1v, adv, bv;
    a1v.x = w1a;        a1v.y = w1b;
    adv.x = w2a - w1a;  adv.y = w2b - w1b;
    bv.x = hb[ca * NPTS + ncol];
    bv.y = hb[(ca + 1) * NPTS + ncol];
    __builtin_prefetch(&hb[(ca + 4) * NPTS + ncol], 0, 3);
    accA = wmma4(a1v, bv, accA);
    accD = wmma4(adv, bv, accD);
  }
  if (C & 3) {  // C==3 tail (conv1 only)
    int ca = C4 + 2 * hi;
    float w1a = (ca     < C) ? w[orow * twoC + ca]         : 0.f;
    float w1b = (ca + 1 < C) ? w[orow * twoC + ca + 1]     : 0.f;
    float w2a = (ca     < C) ? w[orow * twoC + C + ca]     : 0.f;
    float w2b = (ca + 1 < C) ? w[orow * twoC + C + ca + 1] : 0.f;
    v2f a1v, adv, bv;
    a1v.x = w1a;        a1v.y = w1b;
    adv.x = w2a - w1a;  adv.y = w2b - w1b;
    bv.x = (ca     < C) ? hb[ca * NPTS + ncol]       : 0.f;
    bv.y = (ca + 1 < C) ? hb[(ca + 1) * NPTS + ncol] : 0.f;
    accA = wmma4(a1v, bv, accA);
    accD = wmma4(adv, bv, accD);
  }
#pragma unroll
  for (int r = 0; r < 8; r++) {
    int o = obase + r + 8 * hi;
    A[((long)b * 256 + o) * NPTS + ncol] = accA[r];
    D[((long)b * 256 + o) * NPTS + ncol] = accD[r];
  }
}

// y[o,n] = max_k lrelu(s*(A[o,idx[n,k]] + D[o,n]) + b)  -> feats slab
__global__ void edgemax_k(const float* A, const float* D, const int* idx,
                          const float* s, const float* bb, float* yout,
                          int chofs) {
  int b = blockIdx.z, o = blockIdx.y;
  int n = blockIdx.x * blockDim.x + threadIdx.x;
  float dv = D[((long)b * 256 + o) * NPTS + n];
  float sc = s[o], bo = bb[o];
  const float* Ab = A + ((long)b * 256 + o) * NPTS;
  const int* ib = idx + (long)(b * NPTS + n) * KNN;
  float best = -3.4e38f;
#pragma unroll
  for (int k = 0; k < KNN; k++)
    best = fmaxf(best, lrelu(sc * (Ab[ib[k]] + dv) + bo));
  yout[(long)b * FEATC * NPTS + (chofs + o) * NPTS + n] = best;
}

// masked mean-pool to 27 parts via LDS float atomics (+pce on layer 0)
__global__ void partpool_k(const float* y, int chofs, const int* pcol,
                           const float* cnt, const float* pce, int addPce,
                           float* pf) {
  int b = blockIdx.y, c = blockIdx.x, tid = threadIdx.x;
  __shared__ float bins[VP];
  if (tid < VP) bins[tid] = 0.f;
  __syncthreads();
  const float* yb = y + (long)b * FEATC * NPTS + (chofs + c) * NPTS;
  for (int n = tid; n < NPTS; n += blockDim.x)
    atomicAdd(&bins[pcol[b * NPTS + n]], yb[n]);
  __syncthreads();
  if (tid < VP) {
    float v = bins[tid] / cnt[b * 32 + tid];
    if (addPce) v += pce[(b * 64 + c) * 32 + tid];
    pf[((long)b * 256 + c) * 32 + tid] = v;
  }
}

// hop-GCN ec1 decomposition on the 27-vertex graph: A1=W1.pf, D1=(W2-W1).pf
__global__ void hopprep_k(const float* pf, const float* w, int Co, float* A1,
                          float* D1) {
  int b = blockIdx.x, tid = threadIdx.x;
  int twoC = 2 * Co;
  for (int t = tid; t < Co * VP; t += blockDim.x) {
    int o = t / VP, p = t % VP;
    float s1 = 0.f, s2 = 0.f;
    for (int c = 0; c < Co; c++) {
      float w1 = w[o * twoC + c], w2 = w[o * twoC + Co + c];
      float v = pf[((long)b * 256 + c) * 32 + p];
      s1 += w1 * v; s2 += (w2 - w1) * v;
    }
    A1[((long)b * 256 + o) * 32 + p] = s1;
    D1[((long)b * 256 + o) * 32 + p] = s2;
  }
}

__global__ void e1_k(const float* A1, const float* D1, const float* s,
                     const float* bb, float* e1) {
  int b = blockIdx.y, o = blockIdx.x, tid = threadIdx.x;
  float sc = s[o], bo = bb[o];
  const float* Ab = A1 + ((long)b * 256 + o) * 32;
  const float* Db = D1 + ((long)b * 256 + o) * 32;
  float* eb = e1 + ((long)b * 256 + o) * ESTR;
  for (int j = tid; j < VP * VP; j += blockDim.x) {
    int n = j / VP, k = j % VP;
    eb[j] = lrelu(sc * (Ab[k] + Db[n]) + bo);
  }
}

// head attention: logits over k (27), wave32 shuffle softmax, scale e in place
__global__ void att_k(float* e, const float* av, const float* gh, int useGh,
                      int Co) {
  int n = blockIdx.x, hh = blockIdx.y, b = blockIdx.z;
  int k = threadIdx.x;
  int dph = Co / 4;
  bool valid = k < VP;
  int jj = n * VP + (valid ? k : 0);
  float ghv = useGh ? gh[b * ESTR + jj] : 1.f;
  float logit = -3.0e30f;
  if (valid) {
    float acc = 0.f;
    for (int d = 0; d < dph; d++) {
      float v = e[((long)b * 256 + 4 * d + hh) * ESTR + jj];
      acc += av[d] * lrelu(useGh ? ghv * v : v);
    }
    logit = acc;
  }
  float mx = logit;
  for (int off = 16; off; off >>= 1) mx = fmaxf(mx, __shfl_xor(mx, off, 32));
  float ex = valid ? __expf(logit - mx) : 0.f;
  float sm = ex;
  for (int off = 16; off; off >>= 1) sm += __shfl_xor(sm, off, 32);
  float wgt = ex / sm;
  if (valid)
    for (int d = 0; d < dph; d++)
      e[((long)b * 256 + 4 * d + hh) * ESTR + jj] *= wgt;
}

// generic per-batch WMMA GEMM over 729 pixels with conv-bn-lrelu epilogue.
// OOB tile rows/cols use clamped (in-bounds) addresses; their results are
// simply never stored, so the K-loop is branch-free.
__global__ void gemmcbr_k(const float* in, long inBS, const float* w,
                          const float* s, const float* bb, int Cout, int Kin,
                          float* out, long outBS) {
  int b = blockIdx.z;
  int obase = blockIdx.y * 16, jbase = blockIdx.x * 16;
  int lane = threadIdx.x, m16 = lane & 15, hi = lane >> 4;
  const float* inb = in + (long)b * inBS;
  v8f acc = {};
  int orow = obase + m16;
  int orc = orow < Cout ? orow : Cout - 1;   // clamped A row
  int j = jbase + m16;
  bool jok = j < VP * VP;
  int jc = jok ? j : VP * VP - 1;            // clamped B col
#pragma unroll 2
  for (int c0 = 0; c0 < Kin; c0 += 4) {
    int ca = c0 + 2 * hi;
    v2f a, bv;
    a.x = w[orc * Kin + ca];
    a.y = w[orc * Kin + ca + 1];
    bv.x = inb[(long)ca * ESTR + jc];
    bv.y = inb[(long)(ca + 1) * ESTR + jc];
    acc = wmma4(a, bv, acc);
  }
#pragma unroll
  for (int r = 0; r < 8; r++) {
    int o = obase + r + 8 * hi;
    if (o < Cout && jok)
      out[(long)b * outBS + (long)o * ESTR + j] = lrelu(s[o] * acc[r] + bb[o]);
  }
}

// argmax over 5 hop classes -> gauss weights; also emit logits output
__global__ void gh_k(const float* hoplog, float* gh, float* outlog) {
  int b = blockIdx.x, tid = threadIdx.x;
  for (int j = tid; j < VP * VP; j += blockDim.x) {
    float best = -3.4e38f; int arg = 0;
    for (int c = 0; c < 5; c++) {
      float v = hoplog[((long)b * 5 + c) * ESTR + j];
      outlog[((long)b * 5 + c) * VP * VP + j] = v;
      if (v > best) { best = v; arg = c; }
    }
    float a = (float)arg;  // gauss(h), sigma^2 = 2
    gh[b * ESTR + j] = __expf(-a * a * 0.25f) * 0.2820947918f;
  }
}

__global__ void g_k(const float* e2, float* g, int Co) {
  int b = blockIdx.x, tid = threadIdx.x;
  for (int t = tid; t < Co * VP; t += blockDim.x) {
    int o = t / VP, n = t % VP;
    const float* eb = e2 + ((long)b * 256 + o) * ESTR + n * VP;
    float s = 0.f;
    for (int k = 0; k < VP; k++) s += eb[k];
    g[((long)b * 256 + o) * 32 + n] = s * (1.0f / 27.0f);
  }
}

__global__ void p2p_k(const float* g, const int* pcol, float* yout, int chofs,
                      int Co) {
  int b = blockIdx.z, o = blockIdx.y;
  int n = blockIdx.x * blockDim.x + threadIdx.x;
  int p = pcol[b * NPTS + n];
  yout[(long)b * FEATC * NPTS + (chofs + o) * NPTS + n] +=
      g[((long)b * 256 + o) * 32 + p];
}

// ---------------------------------------------------------------------------
extern "C" void kernel_launch(void* const* d_in, const int* in_sizes, int n_in,
                              void* d_out, int out_size, void* d_ws,
                              size_t ws_size, hipStream_t stream) {
  const float* x = (const float*)d_in[0];
  const int* p2v = (const int*)d_in[1];
  const int* prand = (const int*)d_in[2];
  float* out = (float*)d_out;
  float* wsf = (float*)d_ws;

  // workspace layout (floats)
  float* colxx = wsf;                  // 16384
  float* cnt   = colxx + 16384;        // 512
  float* pce   = cnt + 512;            // 32768
  float* pf    = pce + 32768;          // 131072
  float* A1    = pf + 131072;          // 131072
  float* D1    = A1 + 131072;          // 131072
  float* g     = D1 + 131072;          // 131072
  float* gh    = g + 131072;           // 11776
  float* hlog  = gh + 11776;           // 58880
  float* Abuf  = hlog + 58880;         // 4194304
  float* Dbuf  = Abuf + 4194304;       // 4194304
  float* e1    = Dbuf + 4194304;       // 3014656
  float* e2    = e1 + 3014656;         // 3014656
  float* tmp1  = e2 + 3014656;         // 1507328
  int* pcol = (int*)(tmp1 + 1507328);  // 16384 ints
  int* idx = pcol + 16384;             // 327680 ints

  init_parts_k<<<BATCH, 32, 0, stream>>>(p2v, prand, pcol, cnt);
  pce_k<<<BATCH, 32, 0, stream>>>(x, pcol, cnt, (const float*)d_in[15],
                                  (const float*)d_in[16],
                                  (const float*)d_in[17], pce);

  const int Cin[4]  = {3, 64, 64, 128};
  const int Cout[4] = {64, 64, 128, 256};
  const int ofs[4]  = {0, 64, 128, 256};

  const float* hbase = x;
  long hbs = 3L * NPTS;
  for (int L = 0; L < 4; L++) {
    int C = Cin[L], Co = Cout[L];
    const float* cw = (const float*)d_in[3 + 3 * L];
    const float* cs = (const float*)d_in[4 + 3 * L];
    const float* cb = (const float*)d_in[5 + 3 * L];
    int hp = 18 + 14 * L;
    const float* ec1w = (const float*)d_in[hp + 0];
    const float* ec1s = (const float*)d_in[hp + 1];
    const float* ec1b = (const float*)d_in[hp + 2];
    const float* hc1w = (const float*)d_in[hp + 3];
    const float* hc1s = (const float*)d_in[hp + 4];
    const float* hc1b = (const float*)d_in[hp + 5];
    const float* hc2w = (const float*)d_in[hp + 6];
    const float* hc2s = (const float*)d_in[hp + 7];
    const float* hc2b = (const float*)d_in[hp + 8];
    const float* ec2w = (const float*)d_in[hp + 9];
    const float* ec2s = (const float*)d_in[hp + 10];
    const float* ec2b = (const float*)d_in[hp + 11];
    const float* a1p  = (const float*)d_in[hp + 12];
    const float* a2p  = (const float*)d_in[hp + 13];

    colxx_k<<<dim3(NPTS / 256, BATCH), 256, 0, stream>>>(hbase, hbs, C, colxx);
    knn_k<<<dim3(NPTS / 16, BATCH), 32, 0, stream>>>(hbase, hbs, C, colxx, idx);
    gemmAD_k<<<dim3(NPTS / 16, Co / 16, BATCH), 32, 0, stream>>>(
        hbase, hbs, C, 2 * C, cw, Abuf, Dbuf);
    edgemax_k<<<dim3(NPTS / 256, Co, BATCH), 256, 0, stream>>>(
        Abuf, Dbuf, idx, cs, cb, out, ofs[L]);
    partpool_k<<<dim3(Co, BATCH), 256, 0, stream>>>(out, ofs[L], pcol, cnt, pce,
                                                    (L == 0) ? 1 : 0, pf);
    hopprep_k<<<BATCH, 256, 0, stream>>>(pf, ec1w, Co, A1, D1);
    e1_k<<<dim3(Co, BATCH), 128, 0, stream>>>(A1, D1, ec1s, ec1b, e1);
    att_k<<<dim3(VP, 4, BATCH), 32, 0, stream>>>(e1, a1p, gh, 0, Co);
    gemmcbr_k<<<dim3(46, (Co / 2 + 15) / 16, BATCH), 32, 0, stream>>>(
        e1, 256L * ESTR, hc1w, hc1s, hc1b, Co / 2, Co, tmp1, 128L * ESTR);
    gemmcbr_k<<<dim3(46, 1, BATCH), 32, 0, stream>>>(
        tmp1, 128L * ESTR, hc2w, hc2s, hc2b, 5, Co / 2, hlog, 5L * ESTR);
    gh_k<<<BATCH, 256, 0, stream>>>(
        hlog, gh, out + (long)BATCH * FEATC * NPTS + (long)L * BATCH * 5 * VP * VP);
    gemmcbr_k<<<dim3(46, Co / 16, BATCH), 32, 0, stream>>>(
        e1, 256L * ESTR, ec2w, ec2s, ec2b, Co, Co, e2, 256L * ESTR);
    att_k<<<dim3(VP, 4, BATCH), 32, 0, stream>>>(e2, a2p, gh, 1, Co);
    g_k<<<BATCH, 256, 0, stream>>>(e2, g, Co);
    p2p_k<<<dim3(NPTS / 256, Co, BATCH), 256, 0, stream>>>(g, pcol, out, ofs[L], Co);

    hbase = out + (long)ofs[L] * NPTS;  // next layer consumes this slab
    hbs = (long)FEATC * NPTS;
  }
}